// TWKOR_49538152792609
// MI455X (gfx1250) — compile-verified
//
#include <hip/hip_runtime.h>
#include <hip/hip_bf16.h>

#define N_NODES 200000
#define N_EDGES 3200000
#define DIM     64

typedef float v2f __attribute__((ext_vector_type(2)));
typedef float v8f __attribute__((ext_vector_type(8)));

// ---------------------------------------------------------------------------
// Phase 1: side = segment_sum(ego[edge_col] * edge_val, edge_row)
// 16 lanes per edge, float4 per lane -> each edge's 256B row is a coalesced
// gather; ego (51.2MB) and side (51.2MB) both fit in the 192MB L2, so the
// random gather + atomic scatter stay mostly on-chip.
// ---------------------------------------------------------------------------
__global__ void kgat_edge_scatter(const float* __restrict__ ego,
                                  const int*   __restrict__ edge_row,
                                  const int*   __restrict__ edge_col,
                                  const float* __restrict__ edge_val,
                                  float*       __restrict__ side) {
    const int tid  = blockIdx.x * blockDim.x + threadIdx.x;   // < 51.2M, fits int
    const int e    = tid >> 4;                                // edge index
    const int part = tid & 15;                                // 4 floats per part
    if (e >= N_EDGES) return;

    const int   c = edge_col[e];
    const int   r = edge_row[e];
    const float v = edge_val[e];

    const float4 g = *reinterpret_cast<const float4*>(ego + (size_t)c * DIM + part * 4);
    float* dst = side + (size_t)r * DIM + part * 4;
    atomicAdd(dst + 0, g.x * v);
    atomicAdd(dst + 1, g.y * v);
    atomicAdd(dst + 2, g.z * v);
    atomicAdd(dst + 3, g.w * v);
}

// ---------------------------------------------------------------------------
// Phase 2: out = leaky( (ego+side)@W1 + b1 ) + leaky( (ego*side)@W2 + b2 )
// fp32 WMMA: V_WMMA_F32_16X16X4_F32.  One wave32 owns a 16-row stripe and all
// 64 output columns (4 n-tiles x 2 matrices = 8 accumulator tiles).
//
// Layouts (CDNA5 ISA 7.12.2), lane = 32-lane id, l = lane&15, g = lane>>4:
//   A (16x4 f32, 2 VGPR):  row M = l,           a[v] = X[M][kk + 2g + v]
//   B (4x16  f32, 2 VGPR): col N = l,           b[v] = W[kk + 2g + v][N]
//   C/D (16x16 f32, 8 VGPR): elem j -> M = j + 8g, N = l
// ---------------------------------------------------------------------------
__global__ void kgat_fused_gemm(const float* __restrict__ ego,
                                const float* __restrict__ side,
                                const float* __restrict__ W1,
                                const float* __restrict__ b1,
                                const float* __restrict__ W2,
                                const float* __restrict__ b2,
                                float*       __restrict__ out) {
    const int wave = blockIdx.x * (blockDim.x >> 5) + (threadIdx.x >> 5);
    const int lane = threadIdx.x & 31;
    const int l    = lane & 15;
    const int g    = lane >> 4;
    const int m0   = wave * 16;          // first node row of this wave's stripe
    const int arow = m0 + l;             // A-matrix row held by this lane

    v8f acc1[4], acc2[4];
#pragma unroll
    for (int nt = 0; nt < 4; ++nt) {
        acc1[nt] = (v8f){0.f, 0.f, 0.f, 0.f, 0.f, 0.f, 0.f, 0.f};
        acc2[nt] = (v8f){0.f, 0.f, 0.f, 0.f, 0.f, 0.f, 0.f, 0.f};
    }

#pragma unroll 4
    for (int kk = 0; kk < DIM; kk += 4) {
        const int kc = kk + 2 * g;
        const v2f e = *reinterpret_cast<const v2f*>(ego  + (size_t)arow * DIM + kc);
        const v2f s = *reinterpret_cast<const v2f*>(side + (size_t)arow * DIM + kc);
        const v2f a1 = e + s;            // (ego + side) fragment
        const v2f a2 = e * s;            // (ego * side) fragment

#pragma unroll
        for (int nt = 0; nt < 4; ++nt) {
            const int n = nt * 16 + l;
            v2f bw1, bw2;
            bw1.x = W1[(size_t)(kc + 0) * DIM + n];
            bw1.y = W1[(size_t)(kc + 1) * DIM + n];
            bw2.x = W2[(size_t)(kc + 0) * DIM + n];
            bw2.y = W2[(size_t)(kc + 1) * DIM + n];

            acc1[nt] = __builtin_amdgcn_wmma_f32_16x16x4_f32(
                false, a1, false, bw1, (short)0, acc1[nt], false, false);
            acc2[nt] = __builtin_amdgcn_wmma_f32_16x16x4_f32(
                false, a2, false, bw2, (short)0, acc2[nt], false, false);
        }
    }

    // Epilogue: bias + LeakyReLU(0.01) + sum, straight from C/D layout.
#pragma unroll
    for (int nt = 0; nt < 4; ++nt) {
        const int   n   = nt * 16 + l;
        const float bb1 = b1[n];
        const float bb2 = b2[n];
#pragma unroll
        for (int j = 0; j < 8; ++j) {
            const int m  = m0 + j + 8 * g;
            float x1 = acc1[nt][j] + bb1;
            float x2 = acc2[nt][j] + bb2;
            x1 = (x1 > 0.f) ? x1 : 0.01f * x1;
            x2 = (x2 > 0.f) ? x2 : 0.01f * x2;
            out[(size_t)m * DIM + n] = x1 + x2;
        }
    }
}

extern "C" void kernel_launch(void* const* d_in, const int* in_sizes, int n_in,
                              void* d_out, int out_size, void* d_ws, size_t ws_size,
                              hipStream_t stream) {
    const float* ego      = (const float*)d_in[0];
    const float* W1       = (const float*)d_in[1];
    const float* b1       = (const float*)d_in[2];
    const float* W2       = (const float*)d_in[3];
    const float* b2       = (const float*)d_in[4];
    const int*   edge_row = (const int*)d_in[5];
    const int*   edge_col = (const int*)d_in[6];
    const float* edge_val = (const float*)d_in[7];
    float*       out      = (float*)d_out;

    float* side = (float*)d_ws;                 // N_NODES * DIM fp32 = 51.2 MB

    // Zero the segment-sum accumulator (graph-capture-legal async memset).
    hipMemsetAsync(side, 0, (size_t)N_NODES * DIM * sizeof(float), stream);

    // Phase 1: gather + atomic segment-sum. 16 threads/edge.
    {
        const int threads = 256;
        const int total   = N_EDGES * 16;
        const int blocks  = (total + threads - 1) / threads;   // 200000
        kgat_edge_scatter<<<blocks, threads, 0, stream>>>(
            ego, edge_row, edge_col, edge_val, side);
    }

    // Phase 2: fused dual-GEMM + bias + LeakyReLU + add via fp32 WMMA.
    // 128 threads = 4 waves/block, 16 rows/wave -> 64 rows/block, 3125 blocks.
    {
        const int threads = 128;
        const int blocks  = N_NODES / 64;                      // 3125 exact
        kgat_fused_gemm<<<blocks, threads, 0, stream>>>(
            ego, side, W1, b1, W2, b2, out);
    }
}